// DifferentiableParallelBeamRadon_87411174409140
// MI455X (gfx1250) — compile-verified
//
#include <hip/hip_runtime.h>

// ---------------------------------------------------------------------------
// Parallel-beam Radon forward projection, MI455X (gfx1250) version.
//
// Strategy: gather-bound kernel (~47M bilinear samples, ~2MB HBM footprint).
// Stage the whole 256x256 image of one batch into LDS (263KB of the 320KB/WGP)
// via the Tensor Data Mover with pad_enable giving a 257-dword row stride
// (bank-conflict-free column walks at angles near 90 deg), then do all
// bilinear gathers from LDS. The per-detector row-sum reduction + pixel_scale
// is done by V_WMMA_F32_16X16X4_F32 with a constant all-pixel_scale A matrix
// (layout-agnostic because A is uniform).
// ---------------------------------------------------------------------------

#define IMGN   256
#define NANG   180
#define NDET   256
#define LSTR   257            // padded LDS row stride in dwords
#define APW    4              // angles per workgroup
#define BLOCK  512            // 16 waves (wave32)

typedef __attribute__((ext_vector_type(2)))  float        v2f;
typedef __attribute__((ext_vector_type(8)))  float        v8f;
typedef __attribute__((ext_vector_type(4)))  unsigned int u32x4;
typedef __attribute__((ext_vector_type(8)))  int          i32x8;
typedef __attribute__((ext_vector_type(4)))  int          i32x4;

__device__ __forceinline__ float bilinear_lds(const float* L, float ix, float iy) {
  float xf = floorf(ix), yf = floorf(iy);
  float wx1 = ix - xf, wy1 = iy - yf;
  float wx0 = 1.0f - wx1, wy0 = 1.0f - wy1;
  int x0 = (int)xf, y0 = (int)yf;
  int x1 = x0 + 1, y1 = y0 + 1;
  bool vx0 = (unsigned)x0 < 256u;
  bool vx1 = (unsigned)x1 < 256u;
  bool vy0 = (unsigned)y0 < 256u;
  bool vy1 = (unsigned)y1 < 256u;
  int cx0 = vx0 ? x0 : 0, cx1 = vx1 ? x1 : 0;
  int cy0 = vy0 ? y0 : 0, cy1 = vy1 ? y1 : 0;
  float a00 = L[cy0 * LSTR + cx0];
  float a01 = L[cy0 * LSTR + cx1];
  float a10 = L[cy1 * LSTR + cx0];
  float a11 = L[cy1 * LSTR + cx1];
  float wx0m = vx0 ? wx0 : 0.0f;
  float wx1m = vx1 ? wx1 : 0.0f;
  float r0 = wx0m * a00 + wx1m * a01;
  float r1 = wx0m * a10 + wx1m * a11;
  float wy0m = vy0 ? wy0 : 0.0f;
  float wy1m = vy1 ? wy1 : 0.0f;
  return wy0m * r0 + wy1m * r1;
}

__global__ __launch_bounds__(BLOCK)
void radon_fwd_kernel(const float* __restrict__ img, float* __restrict__ out) {
  // Padded image tile: 256 rows x 257 dwords = 263168 B (<= 320KB WGP LDS)
  __shared__ float L[IMGN * LSTR];

  const int b    = blockIdx.x;   // batch
  const int ablk = blockIdx.y;   // block of APW angles
  const float* src = img + (size_t)b * IMGN * IMGN;

#if __has_builtin(__builtin_amdgcn_tensor_load_to_lds) && __has_builtin(__builtin_amdgcn_s_wait_tensorcnt)
  // --- TDM path: one wave issues a single async tensor DMA of the whole
  // 256x256 f32 tile into LDS, padding 1 dword every 256 dwords (stride 257).
  if (threadIdx.x < 32) {
    unsigned long long ga = (unsigned long long)(const void*)src;
    unsigned ldsoff = (unsigned)(unsigned long long)(const void*)&L[0];

    u32x4 g0;
    g0[0] = 1u;                                   // count=1 (valid), no gather
    g0[1] = ldsoff;                               // lds_addr
    g0[2] = (unsigned)(ga & 0xFFFFFFFFull);       // global_addr[31:0]
    g0[3] = (unsigned)((ga >> 32) & 0x1FFFFFFull) // global_addr[56:32]
          | (2u << 30);                           // type = 2 ("image")

    i32x8 g1;
    g1[0] = (2 << 16)       // data_size = 4 bytes
          | (1 << 20)       // pad_enable
          | (7 << 22)       // pad_interval: 256 dwords
          | (0 << 25);      // pad_amount: 1 dword  -> 257-dword LDS stride
    g1[1] = (256 << 16);    // tensor_dim0[15:0] in [31:16]
    g1[2] = (256 << 16);    // tensor_dim1[15:0] in [31:16]
    g1[3] = (256 << 16);    // tile_dim0 in [31:16]
    g1[4] = 256;            // tile_dim1 (tile_dim2 = 0 -> 2D)
    g1[5] = 256;            // tensor_dim0_stride[31:0]
    g1[6] = 0;
    g1[7] = 0;

    i32x4 gz4;
    gz4[0] = 0; gz4[1] = 0; gz4[2] = 0; gz4[3] = 0;
    i32x8 gz8;
    for (int q = 0; q < 8; ++q) gz8[q] = 0;

    // 6-arg form (this toolchain): groups 0..3 + extra group + cache policy.
    __builtin_amdgcn_tensor_load_to_lds(g0, g1, gz4, gz4, gz8, 0);
    __builtin_amdgcn_s_wait_tensorcnt(0);
  }
  __syncthreads();
#else
  // --- Fallback: cooperative load, coalesced global reads, stride-257 stores.
  for (int idx = threadIdx.x; idx < IMGN * IMGN; idx += BLOCK) {
    int y = idx >> 8, x = idx & 255;
    L[y * LSTR + x] = src[idx];
  }
  __syncthreads();
#endif

  const int lane   = threadIdx.x & 31;
  const int wave   = threadIdx.x >> 5;
  const int n      = lane & 15;   // detector within tile == WMMA column N
  const int hi     = lane >> 4;   // lane half selects i-quarter pair
  const int nwaves = BLOCK / 32;
  const float PIXSCALE = 2.0f / 255.0f;

  // 64 tiles per WG: APW angles x 16 detector-tiles of 16.
  for (int t = wave; t < APW * (NDET / 16); t += nwaves) {
    const int al    = t >> 4;
    const int a     = ablk * APW + al;
    const int dbase = (t & 15) << 4;

    const float ang = (float)a * 0.017453292519943295f; // deg2rad
    const float cs  = cosf(ang);
    const float sn  = sinf(ang);

    // ix = cs*(d-127.5) + sn*(i-127.5) + 127.5 ; iy = -sn*(d-127.5) + cs*(i-127.5) + 127.5
    const float dx = (float)(dbase + n) - 127.5f;
    // This lane's two B elements accumulate i-quarters k0=hi and k1=2+hi.
    const float i0 = (float)(hi * 64)       - 127.5f;
    const float i1 = (float)((2 + hi) * 64) - 127.5f;
    float ix0 =  cs * dx + sn * i0 + 127.5f;
    float iy0 = -sn * dx + cs * i0 + 127.5f;
    float ix1 =  cs * dx + sn * i1 + 127.5f;
    float iy1 = -sn * dx + cs * i1 + 127.5f;

    float s0 = 0.0f, s1 = 0.0f;
    for (int j = 0; j < 64; ++j) {
      s0 += bilinear_lds(L, ix0, iy0);
      s1 += bilinear_lds(L, ix1, iy1);
      ix0 += sn; iy0 += cs;   // step i by 1
      ix1 += sn; iy1 += cs;
    }

    float result;
#if __has_builtin(__builtin_amdgcn_wmma_f32_16x16x4_f32)
    // Reduction: D = A x B, A = all-pixel_scale 16x4 f32 (uniform => the
    // k-slot permutation across lanes/VGPRs is irrelevant). Column N = lane%16
    // = detector. D[0,n] = pixel_scale * sum_k B[k,n] = sinogram value.
    v2f amat; amat[0] = PIXSCALE; amat[1] = PIXSCALE;
    v2f bmat; bmat[0] = s0;       bmat[1] = s1;
    v8f czero;
    for (int q = 0; q < 8; ++q) czero[q] = 0.0f;
    v8f dres = __builtin_amdgcn_wmma_f32_16x16x4_f32(
        /*neg_a=*/false, amat, /*neg_b=*/false, bmat,
        /*c_mod=*/(short)0, czero, /*reuse_a=*/false, /*reuse_b=*/false);
    result = dres[0];
#else
    // Cross-lane fallback: lane l and l^16 hold complementary i-quarters.
    float sall = s0 + s1;
    sall += __shfl_xor(sall, 16, 32);
    result = PIXSCALE * sall;
#endif

    if (hi == 0) {
      out[((size_t)b * NANG + (size_t)a) * NDET + (size_t)(dbase + n)] = result;
    }
  }
}

extern "C" void kernel_launch(void* const* d_in, const int* in_sizes, int n_in,
                              void* d_out, int out_size, void* d_ws, size_t ws_size,
                              hipStream_t stream) {
  (void)n_in; (void)out_size; (void)d_ws; (void)ws_size;
  const float* img = (const float*)d_in[0];
  float* out = (float*)d_out;
  const int batch = in_sizes[0] / (IMGN * IMGN);   // = 4
  dim3 grid((unsigned)batch, NANG / APW);          // 4 x 45 workgroups
  radon_fwd_kernel<<<grid, BLOCK, 0, stream>>>(img, out);
}